// SPADE_22883585753561
// MI455X (gfx1250) — compile-verified
//
#include <hip/hip_runtime.h>
#include <math.h>

typedef __attribute__((ext_vector_type(2))) float v2f;
typedef __attribute__((ext_vector_type(8))) float v8f;

#define IMAGE_SIZE 224
#define BLUR_RADIUS 8
#define BLUR_SIGMA  4.0f

// ---------------------------------------------------------------------------
// Kernel 1: dists[n] = ||z_lib[n] - z||.  Uses V_WMMA_F32_16X16X4_F32 for the
// cross term a.z (fp32 in / fp32 acc, no precision loss).  One wave handles a
// tile of 16 library rows; A = z broadcast to 16 rows (16x4), B = z_lib chunk
// (4x16), so D's row 0 holds the 16 dot products.
// A-matrix f32 16x4 layout (ISA 7.12.2): lanes 0-15 supply K=0(.x),1(.y);
// lanes 16-31 supply K=2(.x),3(.y).  B mirrors with N striped across lanes.
// ---------------------------------------------------------------------------
__global__ __launch_bounds__(32) void dist_wmma_kernel(
    const float* __restrict__ z, const float* __restrict__ zlib,
    float* __restrict__ dists, int N, int D) {
  const int lane = threadIdx.x;          // 0..31, full wave32, EXEC all ones
  const int half = lane >> 4;            // 0: K=0,1   1: K=2,3
  const int l    = lane & 15;
  const int row  = blockIdx.x * 16 + l;
  const int rowc = row < N ? row : N - 1;   // clamp (no divergence)
  const float* __restrict__ arow = zlib + (size_t)rowc * D;
  const int kofs = half * 2;

  v8f acc = {};          // dot-product accumulator (C/D of WMMA)
  float ssq = 0.0f;      // this lane's share of sum(a^2)
  float zsq = 0.0f;      // this lane's share of sum(z^2)
  for (int k0 = 0; k0 < D; k0 += 4) {
    float z0 = z[k0 + kofs + 0];
    float z1 = z[k0 + kofs + 1];
    float a0 = arow[k0 + kofs + 0];
    float a1 = arow[k0 + kofs + 1];
    v2f A; A.x = z0; A.y = z1;   // every row of A equals z chunk
    v2f B; B.x = a0; B.y = a1;   // B[k][n] = z_lib[nb+n][k0+k]
    ssq = fmaf(a0, a0, fmaf(a1, a1, ssq));
    zsq = fmaf(z0, z0, fmaf(z1, z1, zsq));
    acc = __builtin_amdgcn_wmma_f32_16x16x4_f32(
        /*neg_a=*/false, A, /*neg_b=*/false, B,
        /*c_mod=*/(short)0, acc, /*reuse_a=*/false, /*reuse_b=*/false);
  }
  // acc[0]: lanes 0-15 hold D[M=0][N=l] = dot(z, z_lib[nb+l])
  float dot = acc[0];
  // combine K={0,1} and K={2,3} partial square-sums across the lane pair
  float ssq_o = __shfl_xor(ssq, 16, 32);
  float zsq_o = __shfl_xor(zsq, 16, 32);
  float d2 = (ssq + ssq_o) + (zsq + zsq_o) - 2.0f * dot;
  if (half == 0 && row < N) dists[row] = sqrtf(fmaxf(d2, 0.0f));
}

// ---------------------------------------------------------------------------
// Kernel 2: top-k smallest distances (deterministic: ties -> smallest index).
// Single 256-thread block.  Writes z_score to d_out[0], indices to ws.
// ---------------------------------------------------------------------------
__global__ __launch_bounds__(256) void topk_kernel(
    const float* __restrict__ dists, const int* __restrict__ kptr,
    float* __restrict__ zscore_out, int* __restrict__ sel_out, int N) {
  __shared__ float sv[256];
  __shared__ int   si[256];
  __shared__ int   sel[16];
  __shared__ float ssum;
  const int t = threadIdx.x;
  int k = *kptr; if (k > 16) k = 16; if (k < 1) k = 1;
  if (t == 0) ssum = 0.0f;
  for (int it = 0; it < k; ++it) {
    __syncthreads();   // sel[] from previous iteration visible
    float best = 3.402823e38f;
    int   bi   = 0x7fffffff;
    for (int i = t; i < N; i += 256) {
      bool taken = false;
      for (int j = 0; j < it; ++j) taken = taken || (sel[j] == i);
      if (!taken) {
        float v = dists[i];
        if (v < best || (v == best && i < bi)) { best = v; bi = i; }
      }
    }
    sv[t] = best; si[t] = bi;
    __syncthreads();
    for (int s = 128; s > 0; s >>= 1) {
      if (t < s) {
        if (sv[t + s] < sv[t] || (sv[t + s] == sv[t] && si[t + s] < si[t])) {
          sv[t] = sv[t + s]; si[t] = si[t + s];
        }
      }
      __syncthreads();
    }
    if (t == 0) { sel[it] = si[0]; ssum += sv[0]; }
  }
  __syncthreads();
  if (t < k)  sel_out[t] = sel[t];
  if (t == 0) zscore_out[0] = ssum / (float)k;
}

// ---------------------------------------------------------------------------
// Kernel 3: per-pixel channel-wise L2 to each of the k nearest maps, min over
// k.  sqrt(min(s2)) == min over k of the L2 norms.
// ---------------------------------------------------------------------------
__global__ __launch_bounds__(256) void level_dist_kernel(
    const float* __restrict__ fmap, const float* __restrict__ lib,
    const int* __restrict__ sel, const int* __restrict__ kptr,
    float* __restrict__ smin, int C, int HW) {
  const int p = blockIdx.x * blockDim.x + threadIdx.x;
  if (p >= HW) return;
  int k = *kptr; if (k > 16) k = 16; if (k < 1) k = 1;
  float best = 3.402823e38f;
  for (int kk = 0; kk < k; ++kk) {
    const float* __restrict__ lrow = lib + (size_t)sel[kk] * C * HW;
    float s = 0.0f;
    for (int c = 0; c < C; ++c) {
      float d = lrow[(size_t)c * HW + p] - fmap[(size_t)c * HW + p];
      s = fmaf(d, d, s);
    }
    best = fminf(best, s);
  }
  smin[p] = sqrtf(best);
}

// ---------------------------------------------------------------------------
// Kernel 4: bilinear upsample (half-pixel centers, edge clamp — matches
// jax.image.resize bilinear upsampling) of the 3 level maps, summed.
// ---------------------------------------------------------------------------
__device__ __forceinline__ float bilin_sample(const float* __restrict__ img,
                                              int S, int oy, int ox) {
  const float scale = (float)S / (float)IMAGE_SIZE;
  float fy = ((float)oy + 0.5f) * scale - 0.5f;
  float fx = ((float)ox + 0.5f) * scale - 0.5f;
  int y0 = (int)floorf(fy); float wy = fy - (float)y0;
  int x0 = (int)floorf(fx); float wx = fx - (float)x0;
  int y0c = min(max(y0, 0), S - 1);
  int y1c = min(max(y0 + 1, 0), S - 1);
  int x0c = min(max(x0, 0), S - 1);
  int x1c = min(max(x0 + 1, 0), S - 1);
  float v00 = img[y0c * S + x0c], v01 = img[y0c * S + x1c];
  float v10 = img[y1c * S + x0c], v11 = img[y1c * S + x1c];
  float top = v00 + wx * (v01 - v00);
  float bot = v10 + wx * (v11 - v10);
  return top + wy * (bot - top);
}

__global__ __launch_bounds__(256) void upsample_sum_kernel(
    const float* __restrict__ s0, const float* __restrict__ s1,
    const float* __restrict__ s2, float* __restrict__ out) {
  const int p = blockIdx.x * blockDim.x + threadIdx.x;
  if (p >= IMAGE_SIZE * IMAGE_SIZE) return;
  const int oy = p / IMAGE_SIZE, ox = p % IMAGE_SIZE;
  out[p] = bilin_sample(s0, 56, oy, ox) +
           bilin_sample(s1, 28, oy, ox) +
           bilin_sample(s2, 14, oy, ox);
}

// ---------------------------------------------------------------------------
// Kernels 5/6: separable 17-tap Gaussian blur (sigma=4), zero "SAME" padding,
// normalized kernel (weights / sum).
// ---------------------------------------------------------------------------
__device__ __forceinline__ float gauss_w(int t) {
  float x = (float)t / BLUR_SIGMA;
  return __expf(-0.5f * x * x);
}

__global__ __launch_bounds__(256) void blur_h_kernel(
    const float* __restrict__ in, float* __restrict__ out) {
  const int p = blockIdx.x * blockDim.x + threadIdx.x;
  if (p >= IMAGE_SIZE * IMAGE_SIZE) return;
  const int oy = p / IMAGE_SIZE, ox = p % IMAGE_SIZE;
  float wsum = 0.0f, acc = 0.0f;
  for (int t = -BLUR_RADIUS; t <= BLUR_RADIUS; ++t) {
    float w = gauss_w(t);
    wsum += w;
    int x = ox + t;
    if (x >= 0 && x < IMAGE_SIZE) acc = fmaf(w, in[oy * IMAGE_SIZE + x], acc);
  }
  out[p] = acc / wsum;
}

__global__ __launch_bounds__(256) void blur_v_kernel(
    const float* __restrict__ in, float* __restrict__ out) {
  const int p = blockIdx.x * blockDim.x + threadIdx.x;
  if (p >= IMAGE_SIZE * IMAGE_SIZE) return;
  const int oy = p / IMAGE_SIZE, ox = p % IMAGE_SIZE;
  float wsum = 0.0f, acc = 0.0f;
  for (int t = -BLUR_RADIUS; t <= BLUR_RADIUS; ++t) {
    float w = gauss_w(t);
    wsum += w;
    int y = oy + t;
    if (y >= 0 && y < IMAGE_SIZE) acc = fmaf(w, in[y * IMAGE_SIZE + ox], acc);
  }
  out[p] = acc / wsum;
}

// ---------------------------------------------------------------------------
extern "C" void kernel_launch(void* const* d_in, const int* in_sizes, int n_in,
                              void* d_out, int out_size, void* d_ws, size_t ws_size,
                              hipStream_t stream) {
  const float* z     = (const float*)d_in[0];   // [1,512]
  const float* fmap0 = (const float*)d_in[1];   // [1,64,56,56]
  const float* fmap1 = (const float*)d_in[2];   // [1,128,28,28]
  const float* fmap2 = (const float*)d_in[3];   // [1,256,14,14]
  const float* z_lib = (const float*)d_in[4];   // [N,512]
  const float* lib0  = (const float*)d_in[5];   // [N,64,56,56]
  const float* lib1  = (const float*)d_in[6];   // [N,128,28,28]
  const float* lib2  = (const float*)d_in[7];   // [N,256,14,14]
  const int*   kptr  = (const int*)d_in[8];     // scalar k (device)

  const int D = in_sizes[0];                    // 512
  const int N = in_sizes[4] / D;                // 1000

  float* out = (float*)d_out;                   // [1 + 224*224]
  float* ws  = (float*)d_ws;

  // workspace layout (float offsets)
  float* dists = ws + 0;                        // [N] (<=1024)
  int*   sel   = (int*)(ws + 1024);             // [16]
  float* s0    = ws + 1040;                     // [56*56]
  float* s1    = s0 + 56 * 56;                  // [28*28]
  float* s2    = s1 + 28 * 28;                  // [14*14]
  float* sraw  = s2 + 14 * 14;                  // [224*224]
  float* stmp  = sraw + IMAGE_SIZE * IMAGE_SIZE;// [224*224]

  // 1) distances via fp32 WMMA GEMV
  dist_wmma_kernel<<<(N + 15) / 16, 32, 0, stream>>>(z, z_lib, dists, N, D);

  // 2) top-k + z_score -> d_out[0]
  topk_kernel<<<1, 256, 0, stream>>>(dists, kptr, out, sel, N);

  // 3) per-level min channel-L2 maps
  level_dist_kernel<<<(56 * 56 + 255) / 256, 256, 0, stream>>>(
      fmap0, lib0, sel, kptr, s0, 64, 56 * 56);
  level_dist_kernel<<<(28 * 28 + 255) / 256, 256, 0, stream>>>(
      fmap1, lib1, sel, kptr, s1, 128, 28 * 28);
  level_dist_kernel<<<(14 * 14 + 255) / 256, 256, 0, stream>>>(
      fmap2, lib2, sel, kptr, s2, 256, 14 * 14);

  // 4) bilinear upsample + sum
  const int npix = IMAGE_SIZE * IMAGE_SIZE;
  upsample_sum_kernel<<<(npix + 255) / 256, 256, 0, stream>>>(s0, s1, s2, sraw);

  // 5) separable Gaussian blur -> d_out[1:]
  blur_h_kernel<<<(npix + 255) / 256, 256, 0, stream>>>(sraw, stmp);
  blur_v_kernel<<<(npix + 255) / 256, 256, 0, stream>>>(stmp, out + 1);
}